// GIN_33088428049204
// MI455X (gfx1250) — compile-verified
//
#include <hip/hip_runtime.h>
#include <hip/hip_bf16.h>
#include <math.h>

// ---------------------------------------------------------------------------
// GIN forward for MI455X (gfx1250, wave32).
// Key trick: nn((1+eps)h + agg) with eps=0 and linear first matmul =>
//   (h + segsum(h[src])) @ W1 == h@W1 + segsum((h@W1)[src])
// so we project to HID=32 BEFORE the edge aggregation (4x less edge traffic
// for layer 0; the 12.8MB projected buffer lives entirely in the 192MB L2).
// GEMMs use V_WMMA_F32_16X16X4_F32 (exact fp32, matches reference).
// ---------------------------------------------------------------------------

#define N_NODES_C 100000
#define N_EDGES_C 1600000
#define HID_C 32

typedef __attribute__((ext_vector_type(2))) float v2f;
typedef __attribute__((ext_vector_type(8))) float v8f;

__device__ __forceinline__ v8f wmma_f32_16x16x4(v2f a, v2f b, v8f c) {
  // 8 args: (neg_a, A, neg_b, B, c_mod, C, reuse_a, reuse_b)
  return __builtin_amdgcn_wmma_f32_16x16x4_f32(
      false, a, false, b, (short)0, c, false, false);
}

// --------------------------------------------------------------------------
// P[n,32] = H[n,DIN] @ W[DIN,32]   (no bias; bias folded later)
// One wave -> 16 node rows, two 16x16 N-tiles. Requires n % 16 == 0.
// --------------------------------------------------------------------------
template <int DIN>
__global__ void k_proj(const float* __restrict__ H, const float* __restrict__ W,
                       float* __restrict__ P, int n_nodes) {
  const int lane = threadIdx.x & 31;
  const int wave = (blockIdx.x * blockDim.x + threadIdx.x) >> 5;
  const int row0 = wave * 16;
  if (row0 >= n_nodes) return;                 // wave-uniform exit

  const int r  = lane & 15;                    // A: row in tile, B: column
  const int kg = lane >> 4;                    // K-group (0: K=0,1  1: K=2,3)
  const float* __restrict__ hrow = H + (size_t)(row0 + r) * DIN;

  v8f acc0 = {};
  v8f acc1 = {};
#pragma unroll 4
  for (int k0 = 0; k0 < DIN; k0 += 4) {
    const int ka = k0 + 2 * kg;
    v2f a;  a.x = hrow[ka];                a.y = hrow[ka + 1];
    v2f b0; b0.x = W[ka * 32 + r];         b0.y = W[(ka + 1) * 32 + r];
    v2f b1; b1.x = W[ka * 32 + 16 + r];    b1.y = W[(ka + 1) * 32 + 16 + r];
    acc0 = wmma_f32_16x16x4(a, b0, acc0);
    acc1 = wmma_f32_16x16x4(a, b1, acc1);
  }
  // D layout: VGPR i -> row = i + 8*(lane>=16), col = lane&15
  const int col = lane & 15;
  const int mhi = (lane >> 4) * 8;
#pragma unroll
  for (int i = 0; i < 8; ++i) {
    const size_t m = (size_t)(row0 + mhi + i) * 32;
    P[m + col]      = acc0[i];
    P[m + 16 + col] = acc1[i];
  }
}

// --------------------------------------------------------------------------
// Zero-fill helper (graph-capture safe; no hipMemsetAsync dependence)
// --------------------------------------------------------------------------
__global__ void k_zero(float* __restrict__ p, int n) {
  int i = blockIdx.x * blockDim.x + threadIdx.x;
  if (i < n) p[i] = 0.0f;
}

// --------------------------------------------------------------------------
// agg[dst,:] += P[src,:]  : one lane per feature -> coalesced 128B gather
// and 128B of f32 atomics per edge (L2-resident destination).
// --------------------------------------------------------------------------
__global__ void k_agg(const long long* __restrict__ ei,
                      const float* __restrict__ P, float* __restrict__ agg,
                      int n_edges) {
  const int gid = blockIdx.x * blockDim.x + threadIdx.x;
  const int e = gid >> 5;
  if (e >= n_edges) return;
  const int f = gid & 31;
  const long long s = ei[e];
  const long long d = ei[(size_t)n_edges + e];
  atomicAdd(&agg[(size_t)d * 32 + f], P[(size_t)s * 32 + f]);
}

// --------------------------------------------------------------------------
// M = relu( relu(P + agg + B1) @ W2 + B2 );  accumulate BN sum / sumsq.
// One wave per 16 rows; block-level LDS reduction -> 64 global atomics/block.
// --------------------------------------------------------------------------
__global__ void k_mlp(const float* __restrict__ P, const float* __restrict__ agg,
                      const float* __restrict__ B1, const float* __restrict__ W2,
                      const float* __restrict__ B2, float* __restrict__ M,
                      float* __restrict__ bnacc, int n_nodes) {
  __shared__ float s_sum[32];
  __shared__ float s_sq[32];
  if (threadIdx.x < 32) { s_sum[threadIdx.x] = 0.f; s_sq[threadIdx.x] = 0.f; }
  __syncthreads();

  const int lane = threadIdx.x & 31;
  const int wave = (blockIdx.x * blockDim.x + threadIdx.x) >> 5;
  const int row0 = wave * 16;
  const int r  = lane & 15;
  const int kg = lane >> 4;

  if (row0 < n_nodes) {                         // wave-uniform; no barrier inside
    const size_t rb = (size_t)(row0 + r) * 32;
    v8f acc0 = {};
    v8f acc1 = {};
#pragma unroll
    for (int k0 = 0; k0 < 32; k0 += 4) {
      const int ka = k0 + 2 * kg;
      v2f a;
      a.x = fmaxf(P[rb + ka]     + agg[rb + ka]     + B1[ka],     0.0f);
      a.y = fmaxf(P[rb + ka + 1] + agg[rb + ka + 1] + B1[ka + 1], 0.0f);
      v2f b0; b0.x = W2[ka * 32 + r];      b0.y = W2[(ka + 1) * 32 + r];
      v2f b1; b1.x = W2[ka * 32 + 16 + r]; b1.y = W2[(ka + 1) * 32 + 16 + r];
      acc0 = wmma_f32_16x16x4(a, b0, acc0);
      acc1 = wmma_f32_16x16x4(a, b1, acc1);
    }
    const int col = lane & 15;
    const int mhi = (lane >> 4) * 8;
    const float bb0 = B2[col], bb1 = B2[16 + col];
    float lsum0 = 0.f, lsq0 = 0.f, lsum1 = 0.f, lsq1 = 0.f;
#pragma unroll
    for (int i = 0; i < 8; ++i) {
      const size_t m = (size_t)(row0 + mhi + i) * 32;
      const float v0 = fmaxf(acc0[i] + bb0, 0.0f);
      const float v1 = fmaxf(acc1[i] + bb1, 0.0f);
      M[m + col]      = v0;
      M[m + 16 + col] = v1;
      lsum0 += v0; lsq0 += v0 * v0;
      lsum1 += v1; lsq1 += v1 * v1;
    }
    atomicAdd(&s_sum[col], lsum0);      atomicAdd(&s_sq[col], lsq0);
    atomicAdd(&s_sum[16 + col], lsum1); atomicAdd(&s_sq[16 + col], lsq1);
  }
  __syncthreads();
  if (threadIdx.x < 32) {
    atomicAdd(&bnacc[threadIdx.x], s_sum[threadIdx.x]);
    atomicAdd(&bnacc[32 + threadIdx.x], s_sq[threadIdx.x]);
  }
}

// --------------------------------------------------------------------------
// BN finalize: per-feature scale/shift from (sum, sumsq).
// --------------------------------------------------------------------------
__global__ void k_bnfin(const float* __restrict__ bnacc,
                        const float* __restrict__ gamma,
                        const float* __restrict__ beta,
                        float* __restrict__ scsh, float inv_n) {
  const int c = threadIdx.x;
  if (c < 32) {
    const float mu  = bnacc[c] * inv_n;
    const float var = bnacc[32 + c] * inv_n - mu * mu;
    const float inv = rsqrtf(var + 1e-5f);
    const float sc  = inv * gamma[c];
    scsh[c]      = sc;
    scsh[32 + c] = beta[c] - mu * sc;
  }
}

// --------------------------------------------------------------------------
// BN apply: H = M * scale[c] + shift[c]
// --------------------------------------------------------------------------
__global__ void k_bnapply(const float* __restrict__ M,
                          const float* __restrict__ scsh,
                          float* __restrict__ H, int n_elem) {
  const int i = blockIdx.x * blockDim.x + threadIdx.x;
  if (i < n_elem) {
    const int c = i & 31;
    H[i] = M[i] * scsh[c] + scsh[32 + c];
  }
}

// --------------------------------------------------------------------------
// Head: relu(h@fc1 + b1) @ fc2 + b2 -> log_softmax. ~1.3 KFLOP/node, trivial.
// --------------------------------------------------------------------------
__global__ void k_head(const float* __restrict__ H,
                       const float* __restrict__ fc1w, const float* __restrict__ fc1b,
                       const float* __restrict__ fc2w, const float* __restrict__ fc2b,
                       float* __restrict__ out, int n_nodes) {
  const int i = blockIdx.x * blockDim.x + threadIdx.x;
  if (i >= n_nodes) return;
  const float* __restrict__ h = H + (size_t)i * 32;
  float hv[32];
#pragma unroll
  for (int k = 0; k < 32; ++k) hv[k] = h[k];
  float t[32];
#pragma unroll
  for (int c = 0; c < 32; ++c) {
    float a = fc1b[c];
#pragma unroll
    for (int k = 0; k < 32; ++k) a = fmaf(hv[k], fc1w[k * 32 + c], a);
    t[c] = fmaxf(a, 0.0f);
  }
  float lg[8];
  float mx = -INFINITY;
#pragma unroll
  for (int c = 0; c < 8; ++c) {
    float a = fc2b[c];
#pragma unroll
    for (int k = 0; k < 32; ++k) a = fmaf(t[k], fc2w[k * 8 + c], a);
    lg[c] = a;
    mx = fmaxf(mx, a);
  }
  float se = 0.f;
#pragma unroll
  for (int c = 0; c < 8; ++c) se += expf(lg[c] - mx);
  const float lse = mx + logf(se);
#pragma unroll
  for (int c = 0; c < 8; ++c) out[(size_t)i * 8 + c] = lg[c] - lse;
}

// ---------------------------------------------------------------------------
extern "C" void kernel_launch(void* const* d_in, const int* in_sizes, int n_in,
                              void* d_out, int out_size, void* d_ws, size_t ws_size,
                              hipStream_t stream) {
  (void)in_sizes; (void)n_in; (void)out_size; (void)ws_size;

  const float*     x       = (const float*)d_in[0];
  const long long* ei      = (const long long*)d_in[1];
  const float*     w1_0    = (const float*)d_in[2];
  const float*     b1_0    = (const float*)d_in[3];
  const float*     w1_rest = (const float*)d_in[4];
  const float*     b1_rest = (const float*)d_in[5];
  const float*     w2      = (const float*)d_in[6];
  const float*     b2      = (const float*)d_in[7];
  const float*     gamma   = (const float*)d_in[8];
  const float*     beta    = (const float*)d_in[9];
  const float*     fc1w    = (const float*)d_in[10];
  const float*     fc1b    = (const float*)d_in[11];
  const float*     fc2w    = (const float*)d_in[12];
  const float*     fc2b    = (const float*)d_in[13];
  float*           out     = (float*)d_out;

  const int N = N_NODES_C;
  const int E = N_EDGES_C;
  const size_t nh = (size_t)N * HID_C;           // 3.2M floats

  float* ws    = (float*)d_ws;
  float* P     = ws;                              // [N,32]
  float* agg   = ws + nh;                         // [N,32]
  float* M     = ws + 2 * nh;                     // [N,32]
  float* Hb    = ws + 3 * nh;                     // [N,32]
  float* bnacc = ws + 4 * nh;                     // [64] sum|sumsq
  float* scsh  = bnacc + 64;                      // [64] scale|shift

  const int waveThreads = (N / 16) * 32;          // one wave per 16 rows
  const int gTile = (waveThreads + 255) / 256;    // 782 blocks
  const int gAgg  = ((E * 32) + 255) / 256;       // 200000 blocks
  const int gElem = ((int)nh + 255) / 256;        // 12500 blocks
  const int gNode = (N + 255) / 256;              // 391 blocks

  for (int l = 0; l < 4; ++l) {
    const float* W1 = (l == 0) ? w1_0 : (w1_rest + (size_t)(l - 1) * 32 * 32);
    const float* B1 = (l == 0) ? b1_0 : (b1_rest + (size_t)(l - 1) * 32);

    if (l == 0)
      k_proj<128><<<gTile, 256, 0, stream>>>(x, W1, P, N);
    else
      k_proj<32><<<gTile, 256, 0, stream>>>(Hb, W1, P, N);

    k_zero<<<gElem, 256, 0, stream>>>(agg, (int)nh);
    k_zero<<<1, 64, 0, stream>>>(bnacc, 64);
    k_agg<<<gAgg, 256, 0, stream>>>(ei, P, agg, E);
    k_mlp<<<gTile, 256, 0, stream>>>(P, agg, B1,
                                     w2 + (size_t)l * 32 * 32,
                                     b2 + (size_t)l * 32,
                                     M, bnacc, N);
    k_bnfin<<<1, 32, 0, stream>>>(bnacc, gamma + (size_t)l * 32,
                                  beta + (size_t)l * 32, scsh, 1.0f / (float)N);
    k_bnapply<<<gElem, 256, 0, stream>>>(M, scsh, Hb, (int)nh);
  }
  k_head<<<gNode, 256, 0, stream>>>(Hb, fc1w, fc1b, fc2w, fc2b, out, N);
}